// HungarianMatcher_5531917877926
// MI455X (gfx1250) — compile-verified
//
#include <hip/hip_runtime.h>
#include <hip/hip_bf16.h>
#include <stdint.h>

#define NCLS 32000
#define NT   128
#define NQ   8192

typedef __attribute__((ext_vector_type(2))) float v2f;
typedef __attribute__((ext_vector_type(8))) float v8f;
typedef int v4i __attribute__((ext_vector_type(4)));

#define HAVE_WMMA_F32X4  __has_builtin(__builtin_amdgcn_wmma_f32_16x16x4_f32)
#define HAVE_ASYNC_LDS   __has_builtin(__builtin_amdgcn_global_load_async_to_lds_b128)

// ---------------------------------------------------------------------------
// Exact f32 cross-lane (wave32) sum via the matrix pipe:
// A = per-lane partials (one f32 per lane, second A VGPR = 0), B = ones(4x16).
// D[m][n] = sum_k A[m][k]; summing the 8 D VGPRs in-lane and adding the
// opposite lane-half gives sum over ALL A entries = sum over all 32 lanes,
// independent of the exact A lane mapping. EXEC must be all ones (it is:
// every wave in these kernels is full).
// ---------------------------------------------------------------------------
__device__ __forceinline__ float wave_allsum(float x) {
#if HAVE_WMMA_F32X4
  v2f a; a.x = x;    a.y = 0.0f;
  v2f b; b.x = 1.0f; b.y = 1.0f;
  v8f c = {0.f, 0.f, 0.f, 0.f, 0.f, 0.f, 0.f, 0.f};
  c = __builtin_amdgcn_wmma_f32_16x16x4_f32(false, a, false, b, (short)0, c,
                                            false, false);
  float t = c[0] + c[1] + c[2] + c[3] + c[4] + c[5] + c[6] + c[7];
  t += __shfl_xor(t, 16, 32);
  return t;
#else
  for (int off = 16; off > 0; off >>= 1) x += __shfl_xor(x, off, 32);
  return x;
#endif
}

__device__ __forceinline__ void async_copy16(const float4* g, float4* l) {
#if HAVE_ASYNC_LDS
  typedef v4i __attribute__((address_space(1))) as1_v4i;
  typedef v4i __attribute__((address_space(3))) as3_v4i;
  __builtin_amdgcn_global_load_async_to_lds_b128((as1_v4i*)g, (as3_v4i*)l,
                                                 0, 0);
#else
  *l = *g;
#endif
}

__device__ __forceinline__ void wait_async_lds() {
#if HAVE_ASYNC_LDS
#if __has_builtin(__builtin_amdgcn_s_wait_asynccnt)
  __builtin_amdgcn_s_wait_asynccnt(0);
#else
  asm volatile("s_wait_asynccnt 0" ::: "memory");
#endif
#endif
}

// ---------------------------------------------------------------------------
// Kernel A: single-pass online logsumexp per row (one wave32 per row).
// 1.05 GB streamed once -> HBM roofline (~45 us @ 23.3 TB/s). b128 loads +
// global_prefetch; WMMA matrix-pipe epilogue reduction.
// ---------------------------------------------------------------------------
__global__ __launch_bounds__(256) void hm_row_stats(
    const float* __restrict__ logits, float* __restrict__ rowmax,
    float* __restrict__ rowsum, int nrows) {
  const int gwave = (blockIdx.x * blockDim.x + threadIdx.x) >> 5;
  const int lane  = threadIdx.x & 31;
  if (gwave >= nrows) return;

  const float4* p4 = (const float4*)(logits + (size_t)gwave * NCLS);
  const int nf4 = NCLS / 4;  // 8000, = 250 per lane exactly

  float m = -3.402823466e38f;
  float s = 0.0f;
  for (int k = lane; k < nf4; k += 32) {
    __builtin_prefetch(p4 + k + 256, 0, 3);  // global_prefetch_b8
    float4 x = p4[k];
    {
      float e = x.x, nm = fmaxf(m, e);
      s = s * __expf(m - nm) + __expf(e - nm); m = nm;
    }
    {
      float e = x.y, nm = fmaxf(m, e);
      s = s * __expf(m - nm) + __expf(e - nm); m = nm;
    }
    {
      float e = x.z, nm = fmaxf(m, e);
      s = s * __expf(m - nm) + __expf(e - nm); m = nm;
    }
    {
      float e = x.w, nm = fmaxf(m, e);
      s = s * __expf(m - nm) + __expf(e - nm); m = nm;
    }
  }
  // wave max, then rescale each lane's partial to the common max
  float M = m;
  for (int off = 16; off > 0; off >>= 1) M = fmaxf(M, __shfl_xor(M, off, 32));
  float sp = s * __expf(m - M);
  float S  = wave_allsum(sp);  // v_wmma_f32_16x16x4_f32 reduction
  if (lane == 0) { rowmax[gwave] = M; rowsum[gwave] = S; }
}

// ---------------------------------------------------------------------------
// Kernel B: costT[t][i] = 1 - exp(x[i,tgt_t] - M_i) / S_i   (transposed 128xNQ)
// ---------------------------------------------------------------------------
__global__ __launch_bounds__(NT) void hm_cost(
    const float* __restrict__ logits, const long long* __restrict__ targets,
    const float* __restrict__ rowmax, const float* __restrict__ rowsum,
    float* __restrict__ costT, int nrows) {
  const int i = blockIdx.x;
  const int t = threadIdx.x;
  if (i >= nrows) return;
  const int col = (int)targets[t];
  const float M = rowmax[i];
  const float S = rowsum[i];
  const float logit = logits[(size_t)i * NCLS + col];
  const float p = __expf(logit - M) / S;
  costT[(size_t)t * nrows + i] = 1.0f - p;
}

// ---------------------------------------------------------------------------
// Kernel C: Jonker-Volgenant LSAP on the 128 x 8192 matrix, one workgroup,
// all state in LDS (~130 KB of CDNA5's 320 KB). Thread 0 does scalar
// bookkeeping; 1024 threads do the 8192-wide relax + packed-key argmin.
// Cost rows are staged through LDS with async global->LDS b128 copies.
// ---------------------------------------------------------------------------
__device__ __forceinline__ unsigned long long packKey(float v, int j) {
  unsigned u = __float_as_uint(v);
  u = (u & 0x80000000u) ? ~u : (u | 0x80000000u);  // monotone float->u32
  return ((unsigned long long)u << 32) | (unsigned)j;  // first-min-index ties
}

__global__ __launch_bounds__(1024) void hm_lsap(const float* __restrict__ costT,
                                                float* __restrict__ outbuf) {
  __shared__ float              sh_shortest[NQ];   // 32 KB
  __shared__ float              sh_v[NQ];          // 32 KB
  __shared__ float              sh_row[NQ];        // 32 KB async staging
  __shared__ short              sh_row4col[NQ];    // 16 KB
  __shared__ unsigned char      sh_SC[NQ];         //  8 KB
  __shared__ unsigned char      sh_path[NQ];       //  8 KB
  __shared__ float              sh_u[NT];
  __shared__ unsigned char      sh_SR[NT];
  __shared__ short              sh_col4row[NT];
  __shared__ unsigned long long sh_best;
  __shared__ int                sh_i, sh_sink;
  __shared__ float              sh_minval, sh_ui;

  const int tid = threadIdx.x;
  const int nth = blockDim.x;  // 1024

  for (int j = tid; j < NQ; j += nth) { sh_v[j] = 0.0f; sh_row4col[j] = -1; }
  if (tid < NT) { sh_u[tid] = 0.0f; sh_col4row[tid] = -1; }
  __syncthreads();

  for (int cur = 0; cur < NT; ++cur) {
    for (int j = tid; j < NQ; j += nth) {
      sh_shortest[j] = 3.402823466e38f;
      sh_SC[j] = 0;
    }
    if (tid < NT) sh_SR[tid] = 0;
    if (tid == 0) { sh_i = cur; sh_minval = 0.0f; sh_sink = -1; }
    __syncthreads();

    while (true) {
      const int i = sh_i;
      if (tid == 0) { sh_SR[i] = 1; sh_ui = sh_u[i]; sh_best = ~0ull; }
      __syncthreads();

      // stage cost row i (32 KB) into LDS: async global->LDS, 2 x b128/thread
      const float4* gsrc = (const float4*)(costT + (size_t)i * NQ);
      float4*       ldst = (float4*)sh_row;
      async_copy16(gsrc + tid,        ldst + tid);
      async_copy16(gsrc + tid + nth,  ldst + tid + nth);
      wait_async_lds();
      __syncthreads();

      const float ui = sh_ui;
      const float mv = sh_minval;
      unsigned long long best = ~0ull;
      for (int j = tid; j < NQ; j += nth) {
        if (!sh_SC[j]) {
          float r = mv + sh_row[j] - ui - sh_v[j];
          if (r < sh_shortest[j]) {
            sh_shortest[j] = r;
            sh_path[j] = (unsigned char)i;
          }
          unsigned long long k = packKey(sh_shortest[j], j);
          best = (k < best) ? k : best;
        }
      }
      if (best != ~0ull) atomicMin(&sh_best, best);
      __syncthreads();

      if (tid == 0) {
        int j = (int)(sh_best & 0xFFFFFFFFull);
        sh_minval = sh_shortest[j];
        sh_SC[j] = 1;
        if (sh_row4col[j] < 0) sh_sink = j;
        else                   sh_i = sh_row4col[j];
      }
      __syncthreads();
      if (sh_sink >= 0) break;
    }

    // dual updates (before augmenting, as in the reference)
    const float mv = sh_minval;
    if (tid < NT) {
      if (tid == cur)      sh_u[tid] += mv;
      else if (sh_SR[tid]) sh_u[tid] += mv - sh_shortest[sh_col4row[tid]];
    }
    for (int j = tid; j < NQ; j += nth)
      if (sh_SC[j]) sh_v[j] -= mv - sh_shortest[j];
    __syncthreads();

    // augment alternating path (scalar, <=128 hops)
    if (tid == 0) {
      int j = sh_sink;
      while (true) {
        int i = sh_path[j];
        sh_row4col[j] = (short)i;
        short tmp = sh_col4row[i];
        sh_col4row[i] = (short)j;
        j = tmp;
        if (i == cur) break;
      }
    }
    __syncthreads();
  }

  // Transposed-return convention: rows = col4row sorted ascending; cols follow.
  if (tid < NT) {
    const int myrow = sh_col4row[tid];
    int rank = 0;
    for (int t2 = 0; t2 < NT; ++t2) {
      int r2 = sh_col4row[t2];
      rank += (r2 < myrow) || (r2 == myrow && t2 < tid);
    }
    outbuf[rank]      = (float)myrow;
    outbuf[NT + rank] = (float)tid;
  }
}

// ---------------------------------------------------------------------------
extern "C" void kernel_launch(void* const* d_in, const int* in_sizes, int n_in,
                              void* d_out, int out_size, void* d_ws,
                              size_t ws_size, hipStream_t stream) {
  const float*     logits  = (const float*)d_in[0];
  const long long* targets = (const long long*)d_in[1];
  (void)n_in; (void)out_size; (void)ws_size;

  const int nrows = in_sizes[0] / NCLS;  // 8192

  float* ws     = (float*)d_ws;
  float* rowmax = ws;                 // [nrows]
  float* rowsum = ws + nrows;         // [nrows]
  float* costT  = ws + 2 * nrows;     // [NT * nrows]

  const int blocksA = (nrows * 32 + 255) / 256;  // one wave32 per row
  hm_row_stats<<<blocksA, 256, 0, stream>>>(logits, rowmax, rowsum, nrows);
  hm_cost<<<nrows, NT, 0, stream>>>(logits, targets, rowmax, rowsum, costT,
                                    nrows);
  hm_lsap<<<1, 1024, 0, stream>>>(costT, (float*)d_out);
}